// DeformRoIPooling_63428077027595
// MI455X (gfx1250) — compile-verified
//
#include <hip/hip_runtime.h>

// ---------------------------------------------------------------------------
// Deformable ROI pooling for MI455X (gfx1250, wave32).
//
// Strategy (see analysis): latency/gather-bound op, no GEMM structure ->
// NHWC repack + per-bin wave32 gather with CDNA5 async global->LDS staging
// (double buffered, ASYNCcnt-synchronized) so 4x4 bilinear corner fetches
// are pipelined without VGPR pressure.
// ---------------------------------------------------------------------------

#define C_CH  128
#define H_IN  80
#define W_IN  80
#define OUTS  7     // OUT_SIZE == PART_SIZE == 7  => part index == bin index
#define NBIN  (OUTS * OUTS)

// Low 32 bits of a generic pointer into LDS == LDS byte address (gfx1250
// aperture encoding: addr[31:0] is the LDS offset).
__device__ __forceinline__ unsigned lds_addr_u32(const void* p) {
  return (unsigned)(unsigned long long)p;
}

// CDNA5 async copy: 16B per lane, global -> LDS, tracked by ASYNCcnt.
// vdst (single VGPR) = per-lane LDS byte address, vaddr (VGPR pair) = 64-bit
// global byte address, no SADDR.
#define ASYNC_LOAD_B128(ldsOff, gptr)                                         \
  asm volatile("global_load_async_to_lds_b128 %0, %1, off"                    \
               :: "v"(ldsOff),                                                \
                  "v"((unsigned long long)(gptr))                             \
               : "memory")

#define WAIT_ASYNCCNT_LE4() asm volatile("s_wait_asynccnt 4" ::: "memory")
#define WAIT_ASYNCCNT_0()   asm volatile("s_wait_asynccnt 0" ::: "memory")
#define WAIT_DSCNT_0()      asm volatile("s_wait_dscnt 0"    ::: "memory")

// ---------------------------------------------------------------------------
// Kernel A: NCHW -> NHWC repack through LDS tiles (32 x-positions x 32
// channels, padded to kill bank conflicts). One-time 6.5 MB pass; makes every
// later gather point a contiguous 512B (128ch * f32) read.
// ---------------------------------------------------------------------------
__global__ void nchw_to_nhwc_kernel(const float* __restrict__ src,
                                    float* __restrict__ dst) {
  __shared__ float tile[32][33];
  int t  = blockIdx.x;
  int ct = t & 3;  t >>= 2;      // 4 channel tiles (128/32)
  int xt = t % 3;  t /= 3;       // 3 x tiles (ceil(80/32))
  int y  = t % H_IN;
  int b  = t / H_IN;
  const int tx = threadIdx.x;    // 0..31
  const int ty = threadIdx.y;    // 0..7

  #pragma unroll
  for (int i = ty; i < 32; i += 8) {
    int x = xt * 32 + tx;        // coalesced over x
    int c = ct * 32 + i;
    if (x < W_IN)
      tile[i][tx] = src[(((size_t)b * C_CH + c) * H_IN + y) * W_IN + x];
  }
  __syncthreads();
  #pragma unroll
  for (int i = ty; i < 32; i += 8) {
    int x = xt * 32 + i;
    int c = ct * 32 + tx;        // coalesced over c
    if (x < W_IN)
      dst[(((size_t)b * H_IN + y) * W_IN + x) * C_CH + c] = tile[tx][i];
  }
}

// ---------------------------------------------------------------------------
// Kernel B: one wave32 per (roi, ph, pw) bin; lane l owns channels 4l..4l+3.
// NHWC=true : gather via async global->LDS b128, double-buffered.
// NHWC=false: fallback direct NCHW gather (workspace too small).
// ---------------------------------------------------------------------------
template <bool NHWC>
__global__ void deform_roi_pool_kernel(const float* __restrict__ src,
                                       const float* __restrict__ rois,
                                       const float* __restrict__ offs,
                                       float* __restrict__ out,
                                       int nBins) {
  const int lane = threadIdx.x & 31;
  const int wv   = threadIdx.x >> 5;           // 8 waves / block
  const int bin  = blockIdx.x * 8 + wv;
  if (bin >= nBins) return;

  const int n  = bin / NBIN;
  const int pp = bin - n * NBIN;
  const int ph = pp / OUTS;
  const int pw = pp - ph * OUTS;

  // Wave-uniform ROI geometry (matches jnp.round == rintf half-even).
  const float* r = rois + (size_t)n * 5;
  const int   b   = (int)r[0];
  const float scl = 1.0f / 16.0f;
  const float sw  = rintf(r[1]) * scl - 0.5f;
  const float sh  = rintf(r[2]) * scl - 0.5f;
  const float ew  = (rintf(r[3]) + 1.0f) * scl - 0.5f;
  const float eh  = (rintf(r[4]) + 1.0f) * scl - 0.5f;
  const float roiW = fmaxf(ew - sw, 0.1f);
  const float roiH = fmaxf(eh - sh, 0.1f);
  const float binW = roiW / (float)OUTS;
  const float binH = roiH / (float)OUTS;
  const float subW = binW * 0.25f;
  const float subH = binH * 0.25f;
  const float tx = offs[((size_t)n * 2 + 0) * NBIN + pp] * 0.1f;
  const float ty = offs[((size_t)n * 2 + 1) * NBIN + pp] * 0.1f;
  const float wstart = (float)pw * binW + sw + tx * roiW;
  const float hstart = (float)ph * binH + sh + ty * roiH;

  float acc[4] = {0.f, 0.f, 0.f, 0.f};
  float cnt = 0.f;

  // Sample s = 4*i + j  (i: vertical subsample, j: horizontal).
  auto coords = [&](int s, int& h0, int& w0, int& h1, int& w1,
                    float& lh, float& lw, float& vf) {
    const int i = s >> 2, j = s & 3;
    const float h = hstart + (float)i * subH;
    const float w = wstart + (float)j * subW;
    vf = (w >= -0.5f && w <= (float)W_IN - 0.5f &&
          h >= -0.5f && h <= (float)H_IN - 0.5f) ? 1.0f : 0.0f;
    const float hc = fminf(fmaxf(h, 0.0f), (float)(H_IN - 1));
    const float wc = fminf(fmaxf(w, 0.0f), (float)(W_IN - 1));
    h0 = (int)floorf(hc);
    w0 = (int)floorf(wc);
    h1 = min(h0 + 1, H_IN - 1);
    w1 = min(w0 + 1, W_IN - 1);
    lh = hc - (float)h0;
    lw = wc - (float)w0;
  };

  if constexpr (NHWC) {
    // Per-wave double buffer: [stage][corner][128ch] = 2*4*512B = 4KB/wave.
    __shared__ alignas(16) float smem[8][2][4][C_CH];
    const unsigned ldsBase =
        lds_addr_u32(&smem[wv][0][0][0]) + (unsigned)(lane << 4);

    auto cornerPtr = [&](int y, int x) {
      return src + ((((size_t)b * H_IN + y) * W_IN + x) << 7) + (lane << 2);
    };
    auto issue = [&](int s) {
      int h0, w0, h1, w1; float lh, lw, vf;
      coords(s, h0, w0, h1, w1, lh, lw, vf);
      const unsigned base = ldsBase + (unsigned)((s & 1) * (4 * C_CH * 4));
      ASYNC_LOAD_B128(base +    0u, cornerPtr(h0, w0));
      ASYNC_LOAD_B128(base +  512u, cornerPtr(h0, w1));
      ASYNC_LOAD_B128(base + 1024u, cornerPtr(h1, w0));
      ASYNC_LOAD_B128(base + 1536u, cornerPtr(h1, w1));
    };
    auto consume = [&](int s) {
      int h0, w0, h1, w1; float lh, lw, vf;
      coords(s, h0, w0, h1, w1, lh, lw, vf);
      const float w00 = vf * (1.0f - lh) * (1.0f - lw);
      const float w01 = vf * (1.0f - lh) * lw;
      const float w10 = vf * lh * (1.0f - lw);
      const float w11 = vf * lh * lw;
      const float4* q =
          reinterpret_cast<const float4*>(&smem[wv][s & 1][0][lane << 2]);
      const float4 v00 = q[0];   // corner stride = 128 floats = 32 float4s
      const float4 v01 = q[32];
      const float4 v10 = q[64];
      const float4 v11 = q[96];
      acc[0] += w00 * v00.x + w01 * v01.x + w10 * v10.x + w11 * v11.x;
      acc[1] += w00 * v00.y + w01 * v01.y + w10 * v10.y + w11 * v11.y;
      acc[2] += w00 * v00.z + w01 * v01.z + w10 * v10.z + w11 * v11.z;
      acc[3] += w00 * v00.w + w01 * v01.w + w10 * v10.w + w11 * v11.w;
      cnt += vf;
    };

    // Software pipeline: prefill both stages, then steady state
    //   wait(stage s done) -> consume(s) -> drain LDS reads -> refill stage.
    issue(0);
    issue(1);
    #pragma unroll
    for (int s = 0; s < 16; ++s) {
      if (s < 15) { WAIT_ASYNCCNT_LE4(); } else { WAIT_ASYNCCNT_0(); }
      consume(s);
      if (s + 2 < 16) {
        WAIT_DSCNT_0();   // LDS reads of this stage retired before overwrite
        issue(s + 2);
      }
    }
  } else {
    // Fallback: direct NCHW gather (uncoalesced but correct).
    const float* base = src + (size_t)b * C_CH * H_IN * W_IN;
    #pragma unroll 4
    for (int s = 0; s < 16; ++s) {
      int h0, w0, h1, w1; float lh, lw, vf;
      coords(s, h0, w0, h1, w1, lh, lw, vf);
      const float w00 = vf * (1.0f - lh) * (1.0f - lw);
      const float w01 = vf * (1.0f - lh) * lw;
      const float w10 = vf * lh * (1.0f - lw);
      const float w11 = vf * lh * lw;
      #pragma unroll
      for (int k = 0; k < 4; ++k) {
        const int c = (lane << 2) + k;
        const float* pc = base + (size_t)c * H_IN * W_IN;
        acc[k] += w00 * pc[h0 * W_IN + w0] + w01 * pc[h0 * W_IN + w1] +
                  w10 * pc[h1 * W_IN + w0] + w11 * pc[h1 * W_IN + w1];
      }
      cnt += vf;
    }
  }

  // out[n, c, ph, pw]  (N, 128, 7, 7)
  const float inv = (cnt > 0.0f) ? (1.0f / fmaxf(cnt, 1.0f)) : 0.0f;
  const size_t ob = ((size_t)n * C_CH) * NBIN + (size_t)pp;
  const int c0 = lane << 2;
  #pragma unroll
  for (int k = 0; k < 4; ++k)
    out[ob + (size_t)(c0 + k) * NBIN] = acc[k] * inv;
}

// ---------------------------------------------------------------------------
extern "C" void kernel_launch(void* const* d_in, const int* in_sizes, int n_in,
                              void* d_out, int out_size, void* d_ws,
                              size_t ws_size, hipStream_t stream) {
  const float* data = (const float*)d_in[0];
  const float* rois = (const float*)d_in[1];
  const float* offs = (const float*)d_in[2];
  float* out = (float*)d_out;

  const int N     = in_sizes[1] / 5;
  const int Bn    = in_sizes[0] / (C_CH * H_IN * W_IN);
  const int nBins = N * NBIN;
  const int blocks = (nBins + 7) / 8;        // 8 wave32s per block

  const size_t need = (size_t)in_sizes[0] * sizeof(float);
  if (ws_size >= need) {
    float* ws = (float*)d_ws;
    dim3 blk(32, 8);
    nchw_to_nhwc_kernel<<<Bn * H_IN * 3 * 4, blk, 0, stream>>>(data, ws);
    deform_roi_pool_kernel<true><<<blocks, 256, 0, stream>>>(ws, rois, offs,
                                                             out, nBins);
  } else {
    deform_roi_pool_kernel<false><<<blocks, 256, 0, stream>>>(data, rois, offs,
                                                              out, nBins);
  }
}